// DifferentiableNMS_56736517980682
// MI455X (gfx1250) — compile-verified
//
#include <hip/hip_runtime.h>
#include <math.h>

typedef __attribute__((ext_vector_type(2))) float v2f;
typedef __attribute__((ext_vector_type(8))) float v8f;

// Problem shape (fixed by setup_inputs)
constexpr int kB = 32, kP = 4096, kR = 128, kF = 256;
constexpr float kTempInv = 10.0f;   // 1/TEMPERATURE
constexpr float kEps     = 1e-10f;
constexpr float kThresh  = 0.05f;

// Workspace layout (float offsets)
constexpr size_t OFF_SIM = 0;                               // [B,P,R]  64 MB
constexpr size_t OFF_RN  = OFF_SIM + (size_t)kB * kP * kR;  // [B,P] feature row norms
constexpr size_t OFF_CN  = OFF_RN  + (size_t)kB * kP;       // [B,R] centroid norms
constexpr size_t OFF_ROW = OFF_CN  + (size_t)kB * kR;       // [B,P] sinkhorn row
constexpr size_t OFF_COL = OFF_ROW + (size_t)kB * kP;       // [B,R] sinkhorn col
constexpr size_t OFF_MXD = OFF_COL + (size_t)kB * kR;       // [B]   max dist (uint bits)
constexpr size_t OFF_CTR = OFF_MXD + (size_t)kB;            // [B,P,2] box centers

// fp32 WMMA: D(16x16) += A(16x4) * B(4x16); one wave per tile.
__device__ __forceinline__ v8f wmma_k4(v2f a, v2f b, v8f c) {
  return __builtin_amdgcn_wmma_f32_16x16x4_f32(false, a, false, b, (short)0, c,
                                               false, false);
}

// Async global->LDS 16B copy per lane (CDNA5 TDM-family path, ASYNCcnt).
__device__ __forceinline__ void async_g2l_b128(unsigned lds_byte_off,
                                               unsigned long long gaddr) {
  asm volatile("global_load_async_to_lds_b128 %0, %1, off"
               :: "v"(lds_byte_off), "v"(gaddr) : "memory");
}
__device__ __forceinline__ void wait_asynccnt0() {
  asm volatile("s_wait_asynccnt 0x0" ::: "memory");
}

// ---- L2 row norms: one wave per row of F=256 floats -------------------------
__global__ void dnms_norm_kernel(const float* __restrict__ x,
                                 float* __restrict__ out, int nrows) {
  const int gid  = blockIdx.x * blockDim.x + threadIdx.x;
  const int row  = gid >> 5;
  const int lane = threadIdx.x & 31;
  if (row >= nrows) return;
  const float4* rp = (const float4*)(x + (size_t)row * kF) + lane * 2;
  const float4 a = rp[0];
  const float4 b = rp[1];
  float s = a.x * a.x + a.y * a.y + a.z * a.z + a.w * a.w +
            b.x * b.x + b.y * b.y + b.z * b.z + b.w * b.w;
#pragma unroll
  for (int off = 16; off > 0; off >>= 1) s += __shfl_xor(s, off, 32);
  if (lane == 0) out[row] = fmaxf(sqrtf(s), 1e-12f);
}

__global__ void dnms_maxd_init_kernel(unsigned* __restrict__ mxd) {
  if (threadIdx.x < kB) mxd[threadIdx.x] = 0u;
}

// ---- box centers + per-batch max spatial distance ---------------------------
__global__ void dnms_ctr_kernel(const float* __restrict__ boxes,
                                const float* __restrict__ cs,
                                float* __restrict__ ctr,
                                unsigned* __restrict__ mxd) {
  const int idx = blockIdx.x * blockDim.x + threadIdx.x;  // over B*P
  const int b   = idx >> 12;                              // P = 4096
  const float4 bx = *(const float4*)(boxes + (size_t)idx * 4);
  const float cx = (bx.x + bx.z) * 0.5f;
  const float cy = (bx.y + bx.w) * 0.5f;
  ctr[(size_t)idx * 2 + 0] = cx;
  ctr[(size_t)idx * 2 + 1] = cy;
  const float* c = cs + (size_t)b * kR * 2;
  float m = 0.0f;
  for (int r = 0; r < kR; ++r) {
    const float dx = cx - c[2 * r], dy = cy - c[2 * r + 1];
    m = fmaxf(m, dx * dx + dy * dy);
  }
  m = sqrtf(m);
#pragma unroll
  for (int off = 16; off > 0; off >>= 1) m = fmaxf(m, __shfl_xor(m, off, 32));
  if ((threadIdx.x & 31) == 0) atomicMax(&mxd[b], __float_as_uint(m));
}

// ---- cost matrix + similarity via fp32 WMMA ---------------------------------
// Block = 8 waves, all sharing one 16(p) x 256(F) A tile. The tile is one
// contiguous 16 KB region of `feat`, staged into LDS once via async b128
// copies (row stride padded to 260 floats -> conflict-free ds_load_b64).
constexpr int kAStride = 260;  // 256 + 4 pad floats
__global__ void dnms_sim_kernel(const float* __restrict__ feat,
                                const float* __restrict__ cent,
                                const float* __restrict__ scores,
                                const float* __restrict__ rn,
                                const float* __restrict__ cn,
                                const float* __restrict__ ctr,
                                const float* __restrict__ cs,
                                const unsigned* __restrict__ mxd,
                                float* __restrict__ sim) {
  __shared__ float As[16 * kAStride];
  const int tid  = threadIdx.x;
  const int lane = tid & 31;
  const int wave = tid >> 5;              // 0..7 -> r-tile
  const int b    = blockIdx.x >> 8;       // kP/16 = 256 p-tiles per batch
  const int p0   = (blockIdx.x & 255) << 4;
  const int r0   = wave << 4;
  const int m    = lane & 15;             // A/B row index (M / N)
  const int kb   = (lane >> 4) << 1;      // K base: 0 or 2 per half-wave

  // Stage A: rows p0..p0+15 of feat are contiguous (16 KB = 1024 float4).
  const float* Abase = feat + ((size_t)(b * kP + p0)) * kF;
#pragma unroll
  for (int j = 0; j < 4; ++j) {
    const int i   = tid + (j << 8);       // float4 group index 0..1023
    const int row = i >> 6;               // 64 groups per 256-float row
    const int c4  = i & 63;
    const unsigned lds_off =
        (unsigned)(unsigned long long)&As[row * kAStride + (c4 << 2)];
    async_g2l_b128(lds_off, (unsigned long long)(Abase + ((size_t)i << 2)));
  }
  wait_asynccnt0();
  __syncthreads();

  const float* Ar = &As[m * kAStride + kb];
  const float* Br = cent + ((size_t)(b * kR + r0 + m)) * kF + kb;
  v8f acc = {0.f, 0.f, 0.f, 0.f, 0.f, 0.f, 0.f, 0.f};
#pragma unroll 4
  for (int k = 0; k < kF; k += 4) {
    const v2f a  = *(const v2f*)(Ar + k);      // ds_load_b64, bank-conflict-free
    const v2f bb = *(const v2f*)(Br + k);      // global_load_b64 (L2-resident)
    acc = wmma_k4(a, bb, acc);
  }
  const int r = r0 + m;
  const float cnr = cn[b * kR + r];
  const float csx = cs[((size_t)(b * kR + r)) * 2 + 0];
  const float csy = cs[((size_t)(b * kR + r)) * 2 + 1];
  const float inv_md = 1.0f / fmaxf(__uint_as_float(mxd[b]), kEps);
#pragma unroll
  for (int i = 0; i < 8; ++i) {           // D row M = i + 8*(lane>>4)
    const int p = p0 + i + ((lane >> 4) << 3);
    const float sc  = scores[b * kP + p];
    const float rnp = rn[b * kP + p];
    const float cx  = ctr[((size_t)(b * kP + p)) * 2 + 0];
    const float cy  = ctr[((size_t)(b * kP + p)) * 2 + 1];
    const float dx = cx - csx, dy = cy - csy;
    const float sd   = sqrtf(dx * dx + dy * dy) * inv_md;
    const float fdot = acc[i] / (rnp * cnr);
    const float cost = (1.0f - sc) + (1.0f - fdot) + sd;
    sim[((size_t)(b * kP + p)) * kR + r] = __expf(-cost * kTempInv);
  }
}

__global__ void dnms_init_rc_kernel(float* __restrict__ rowv,
                                    float* __restrict__ colv) {
  const int idx = blockIdx.x * blockDim.x + threadIdx.x;  // B*P threads
  rowv[idx] = 1.0f;
  if (idx < kB * kR) colv[idx] = 1.0f;
}

// ---- Sinkhorn row update: row = 1/(sim @ col + eps) -------------------------
__global__ void dnms_row_kernel(const float* __restrict__ sim,
                                const float* __restrict__ colv,
                                float* __restrict__ rowv) {
  const int idx = blockIdx.x * blockDim.x + threadIdx.x;  // over B*P
  const int b   = idx >> 12;
  const float* s = sim + (size_t)idx * kR;
  const float* c = colv + b * kR;
  float sum = 0.0f;
#pragma unroll 4
  for (int r = 0; r < kR; r += 4) {
    const float4 sv = *(const float4*)(s + r);
    const float4 cv = *(const float4*)(c + r);
    sum += sv.x * cv.x + sv.y * cv.y + sv.z * cv.z + sv.w * cv.w;
  }
  rowv[idx] = 1.0f / (sum + kEps);
}

// ---- Sinkhorn col update: col = 1/(rowT @ sim + eps) ------------------------
__global__ void dnms_col_kernel(const float* __restrict__ sim,
                                const float* __restrict__ rowv,
                                float* __restrict__ colv) {
  __shared__ float red[512];
  const int b  = blockIdx.x;
  const int r  = threadIdx.x & 127;
  const int ch = threadIdx.x >> 7;  // 4 chunks of 1024 proposals
  float sum = 0.0f;
  const int pend = (ch + 1) * 1024;
  for (int p = ch * 1024; p < pend; ++p)
    sum += sim[((size_t)(b * kP + p)) * kR + r] * rowv[b * kP + p];
  red[threadIdx.x] = sum;
  __syncthreads();
  if (ch == 0) {
    sum = red[r] + red[r + 128] + red[r + 256] + red[r + 384];
    colv[b * kR + r] = 1.0f / (sum + kEps);
  }
}

// ---- aggregated scores + boxes, thresholded ---------------------------------
__global__ void dnms_agg_sb_kernel(const float* __restrict__ sim,
                                   const float* __restrict__ rowv,
                                   const float* __restrict__ colv,
                                   const float* __restrict__ scores,
                                   const float* __restrict__ boxes,
                                   float* __restrict__ out_scores,
                                   float* __restrict__ out_boxes) {
  const int b = blockIdx.x;
  const int r = threadIdx.x;  // 128 threads
  const float cv = colv[b * kR + r];
  float ss = 0.f, a0 = 0.f, a1 = 0.f, a2 = 0.f, a3 = 0.f;
  for (int p = 0; p < kP; ++p) {
    const float w  = sim[((size_t)(b * kP + p)) * kR + r] * rowv[b * kP + p] * cv;
    const float sc = scores[b * kP + p];
    const float4 bx = *(const float4*)(boxes + ((size_t)(b * kP + p)) * 4);
    ss += sc * w;
    a0 += bx.x * w; a1 += bx.y * w; a2 += bx.z * w; a3 += bx.w * w;
  }
  const float keep = ss > kThresh ? 1.0f : 0.0f;
  out_scores[b * kR + r] = ss * keep;
  float* ob = out_boxes + ((size_t)(b * kR + r)) * 4;
  ob[0] = a0 * keep; ob[1] = a1 * keep; ob[2] = a2 * keep; ob[3] = a3 * keep;
}

// ---- aggregated features via fp32 WMMA: softT[R,P] x features[P,F] ----------
// Block = 8 waves covering (b, one 16-r tile, 128 of the 256 f). The A operand
// (sim*row for the shared r-tile) is staged into LDS in 64-p chunks and reused
// by all 8 waves, cutting sim-tensor L2 traffic 8x.
__global__ void dnms_agg_feat_kernel(const float* __restrict__ sim,
                                     const float* __restrict__ rowv,
                                     const float* __restrict__ colv,
                                     const float* __restrict__ feat,
                                     const float* __restrict__ out_scores,
                                     float* __restrict__ out_feat) {
  __shared__ float As[64 * 16];  // 4 KB staged A chunk: [p-in-chunk][r-in-tile]
  const int tid  = threadIdx.x;
  const int lane = tid & 31;
  const int wave = tid >> 5;
  const int half = blockIdx.x & 1;           // which 128 of F
  const int rt   = (blockIdx.x >> 1) & 7;    // r-tile
  const int b    = blockIdx.x >> 4;
  const int r0   = rt << 4;
  const int f0   = (half << 7) + (wave << 4);
  const int m    = lane & 15;
  const int kb   = (lane >> 4) << 1;
  const float* simb = sim + (size_t)b * kP * kR;
  const float* rv   = rowv + b * kP;
  const float* fb   = feat + (size_t)b * kP * kF;
  v8f acc = {0.f, 0.f, 0.f, 0.f, 0.f, 0.f, 0.f, 0.f};
  for (int k0 = 0; k0 < kP; k0 += 64) {
    __syncthreads();
#pragma unroll
    for (int j = 0; j < 4; ++j) {          // stage 64x16 scaled A chunk
      const int idx = tid + (j << 8);      // 0..1023
      const int p = idx >> 4, c = idx & 15;
      As[idx] = simb[(size_t)(k0 + p) * kR + r0 + c] * rv[k0 + p];
    }
    __syncthreads();
#pragma unroll 4
    for (int kk = 0; kk < 64; kk += 4) {
      v2f a, bbv;
      a.x = As[(kk + kb) * 16 + m];
      a.y = As[(kk + kb + 1) * 16 + m];
      const size_t pg = (size_t)(k0 + kk + kb);
      bbv.x = fb[pg * kF + f0 + m];
      bbv.y = fb[(pg + 1) * kF + f0 + m];
      acc = wmma_k4(a, bbv, acc);
    }
  }
#pragma unroll
  for (int i = 0; i < 8; ++i) {            // D row M -> r, N -> f
    const int rr = r0 + i + ((lane >> 4) << 3);
    const float s = out_scores[b * kR + rr];           // already masked
    const float keep = s > kThresh ? 1.0f : 0.0f;
    const float cr = colv[b * kR + rr];                // col scale per D row
    out_feat[((size_t)(b * kR + rr)) * kF + f0 + m] = acc[i] * cr * keep;
  }
}

extern "C" void kernel_launch(void* const* d_in, const int* in_sizes, int n_in,
                              void* d_out, int out_size, void* d_ws, size_t ws_size,
                              hipStream_t stream) {
  const float* scores = (const float*)d_in[0];  // [B,P]
  const float* feat   = (const float*)d_in[1];  // [B,P,F]
  const float* boxes  = (const float*)d_in[2];  // [B,P,4]
  const float* cent   = (const float*)d_in[3];  // [B,R,F]
  const float* cs     = (const float*)d_in[4];  // [B,R,2]

  float* ws   = (float*)d_ws;
  float* sim  = ws + OFF_SIM;
  float* rn   = ws + OFF_RN;
  float* cn   = ws + OFF_CN;
  float* rowv = ws + OFF_ROW;
  float* colv = ws + OFF_COL;
  unsigned* mxd = (unsigned*)(ws + OFF_MXD);
  float* ctr  = ws + OFF_CTR;

  float* out_scores = (float*)d_out;                       // [B,R]
  float* out_boxes  = out_scores + kB * kR;                // [B,R,4]
  float* out_feat   = out_boxes + (size_t)kB * kR * 4;     // [B,R,F]

  dnms_norm_kernel<<<(kB * kP) / 8, 256, 0, stream>>>(feat, rn, kB * kP);
  dnms_norm_kernel<<<(kB * kR) / 8, 256, 0, stream>>>(cent, cn, kB * kR);
  dnms_maxd_init_kernel<<<1, 32, 0, stream>>>(mxd);
  dnms_ctr_kernel<<<(kB * kP) / 256, 256, 0, stream>>>(boxes, cs, ctr, mxd);
  dnms_sim_kernel<<<kB * (kP / 16), 256, 0, stream>>>(feat, cent, scores, rn, cn,
                                                      ctr, cs, mxd, sim);
  dnms_init_rc_kernel<<<(kB * kP) / 256, 256, 0, stream>>>(rowv, colv);
  for (int it = 0; it < 10; ++it) {
    dnms_row_kernel<<<(kB * kP) / 256, 256, 0, stream>>>(sim, colv, rowv);
    dnms_col_kernel<<<kB, 512, 0, stream>>>(sim, rowv, colv);
  }
  dnms_agg_sb_kernel<<<kB, kR, 0, stream>>>(sim, rowv, colv, scores, boxes,
                                            out_scores, out_boxes);
  dnms_agg_feat_kernel<<<kB * (kR / 16) * 2, 256, 0, stream>>>(
      sim, rowv, colv, feat, out_scores, out_feat);

  (void)in_sizes; (void)n_in; (void)out_size; (void)ws_size;
}